// DMPNNLayer_30777735643633
// MI455X (gfx1250) — compile-verified
//
#include <hip/hip_runtime.h>
#include <hip/hip_bf16.h>

#define B_ 8
#define N_ 512
#define H_ 64
#define E_ 8

typedef __attribute__((ext_vector_type(2))) float v2f;
typedef __attribute__((ext_vector_type(8))) float v8f;

// ---------------------------------------------------------------------------
// Kernel 1: mask[i*N+j] = (sum_b adj[b,i,j]) > 0  (8 MB stream)
// ---------------------------------------------------------------------------
__global__ void mask_kernel(const int* __restrict__ adj, float* __restrict__ mask) {
    int ij = blockIdx.x * blockDim.x + threadIdx.x;
    int s = 0;
#pragma unroll
    for (int b = 0; b < B_; ++b) s += adj[(size_t)b * N_ * N_ + ij];
    mask[ij] = (s > 0) ? 1.0f : 0.0f;
}

// ---------------------------------------------------------------------------
// Kernel 2/5: out[r,o] = sum_k X[r,k] * W[o*ldw + k] + bias[o]
// rows = B*N = 4096, cols = 64, K = 64.  WMMA f32 16x16x4, 1 tile per wave.
// grid: (4096/16 tiles_r * 4 tiles_o) / 8 waves = 128 blocks of 256.
// ---------------------------------------------------------------------------
__global__ void gemm64_kernel(const float* __restrict__ X, const float* __restrict__ W,
                              int ldw, const float* __restrict__ bias,
                              float* __restrict__ out) {
    int lane = threadIdx.x & 31;
    int wave = threadIdx.x >> 5;
    int tile = blockIdx.x * 8 + wave;
    int r0 = (tile >> 2) * 16;
    int o0 = (tile & 3) * 16;
    int half = lane >> 4;     // 0: K rows 0,1   1: K rows 2,3
    int l    = lane & 15;     // A: row M=l     B: col N=l
    int k2   = half * 2;
    v8f c = {};
#pragma unroll 4
    for (int kk = 0; kk < H_; kk += 4) {
        v2f a, b;
        a.x = X[(size_t)(r0 + l) * H_ + kk + k2 + 0];
        a.y = X[(size_t)(r0 + l) * H_ + kk + k2 + 1];
        b.x = W[(size_t)(o0 + l) * ldw + kk + k2 + 0];
        b.y = W[(size_t)(o0 + l) * ldw + kk + k2 + 1];
        c = __builtin_amdgcn_wmma_f32_16x16x4_f32(false, a, false, b, (short)0, c,
                                                  false, false);
    }
    float bv = bias[o0 + l];
#pragma unroll
    for (int v = 0; v < 8; ++v) {
        int m = v + 8 * half;
        out[(size_t)(r0 + m) * H_ + o0 + l] = c[v] + bv;
    }
}

// ---------------------------------------------------------------------------
// Kernel 3: S[b,j,e] = sum_i mask[i,j] * edge_attr[b,i,j,e]   (64 MB stream)
// Block handles (b, 8 j-values): 64 (j,e) lanes x 4 i-chunks, LDS reduce.
// Consecutive 64 threads read contiguous 256B per i -> fully coalesced.
// ---------------------------------------------------------------------------
__global__ void sred_kernel(const float* __restrict__ edge_attr,
                            const float* __restrict__ mask,
                            float* __restrict__ S) {
    __shared__ float red[256];
    int b  = blockIdx.x >> 6;            // N/8 = 64 j-chunks per batch
    int j0 = (blockIdx.x & 63) * 8;
    int tid = threadIdx.x;
    int je = tid & 63;
    int jl = je >> 3;
    int e  = je & 7;
    int ic = tid >> 6;                   // i-chunk 0..3 (128 i each)
    int j  = j0 + jl;
    const float* ep = edge_attr + (((size_t)b * N_) * N_ + j) * E_ + e;
    float acc = 0.0f;
    int iend = ic * 128 + 128;
    for (int i = ic * 128; i < iend; ++i) {
        __builtin_prefetch(ep + (size_t)(i + 16) * (N_ * E_), 0, 0); // global_prefetch_b8
        acc += mask[i * N_ + j] * ep[(size_t)i * (N_ * E_)];
    }
    red[tid] = acc;
    __syncthreads();
    if (tid < 64) {
        S[((size_t)b * N_ + j) * E_ + e] =
            red[tid] + red[tid + 64] + red[tid + 128] + red[tid + 192];
    }
}

// ---------------------------------------------------------------------------
// Kernel 4: t[b,j,o] = h[b,j,o] + sum_i mask[i,j]*xh[b,i,o] + sum_e S[b,j,e]*We[o,e]
// WMMA f32 16x16x4 over K=i=512 (128 wmma per tile); 8*32*4 = 1024 tiles,
// 8 waves/block -> 128 blocks.  Epilogue fuses edge-message + residual.
// ---------------------------------------------------------------------------
__global__ void msg_kernel(const float* __restrict__ mask, const float* __restrict__ xh,
                           const float* __restrict__ S, const float* __restrict__ W_w,
                           const float* __restrict__ h, float* __restrict__ t) {
    int lane = threadIdx.x & 31;
    int wave = threadIdx.x >> 5;
    int tile = blockIdx.x * 8 + wave;    // 0..1023
    int b   = tile >> 7;                 // 128 tiles per batch
    int rem = tile & 127;
    int jt  = (rem >> 2) * 16;
    int ot  = (rem & 3) * 16;
    int half = lane >> 4;
    int l    = lane & 15;
    int k2   = half * 2;
    v8f c = {};
#pragma unroll 2
    for (int kk = 0; kk < N_; kk += 4) {
        v2f a, bb;
        // A[m,k] = mask[kk+k, jt+m]  (mask^T tile, coalesced across lanes)
        a.x = mask[(size_t)(kk + k2 + 0) * N_ + jt + l];
        a.y = mask[(size_t)(kk + k2 + 1) * N_ + jt + l];
        // B[k,n] = xh[b, kk+k, ot+n]
        bb.x = xh[((size_t)b * N_ + kk + k2 + 0) * H_ + ot + l];
        bb.y = xh[((size_t)b * N_ + kk + k2 + 1) * H_ + ot + l];
        c = __builtin_amdgcn_wmma_f32_16x16x4_f32(false, a, false, bb, (short)0, c,
                                                  false, false);
    }
    int o = ot + l;
    float we[E_];
#pragma unroll
    for (int e = 0; e < E_; ++e) we[e] = W_w[(size_t)o * (H_ + E_) + H_ + e];
#pragma unroll
    for (int v = 0; v < 8; ++v) {
        int j = jt + v + 8 * half;
        float me = 0.0f;
#pragma unroll
        for (int e = 0; e < E_; ++e) me += S[((size_t)b * N_ + j) * E_ + e] * we[e];
        size_t idx = ((size_t)b * N_ + j) * H_ + o;
        t[idx] = c[v] + me + h[idx];
    }
}

// ---------------------------------------------------------------------------
extern "C" void kernel_launch(void* const* d_in, const int* in_sizes, int n_in,
                              void* d_out, int out_size, void* d_ws, size_t ws_size,
                              hipStream_t stream) {
    const float* h   = (const float*)d_in[0];
    const float* ea  = (const float*)d_in[1];
    const int*   adj = (const int*)d_in[2];
    const float* Ww  = (const float*)d_in[3];
    const float* Wb  = (const float*)d_in[4];
    const float* Uw  = (const float*)d_in[5];
    const float* Ub  = (const float*)d_in[6];
    float* out = (float*)d_out;

    float* mask = (float*)d_ws;              // N*N            = 1 MB
    float* xh   = mask + (size_t)N_ * N_;    // B*N*H          = 1 MB
    float* S    = xh + (size_t)B_ * N_ * H_; // B*N*E          = 128 KB
    float* t    = S + (size_t)B_ * N_ * E_;  // B*N*H          = 1 MB

    mask_kernel<<<(N_ * N_) / 256, 256, 0, stream>>>(adj, mask);
    gemm64_kernel<<<128, 256, 0, stream>>>(h, Ww, H_ + E_, Wb, xh);
    sred_kernel<<<B_ * (N_ / 8), 256, 0, stream>>>(ea, mask, S);
    msg_kernel<<<128, 256, 0, stream>>>(mask, xh, S, Ww, h, t);
    gemm64_kernel<<<128, 256, 0, stream>>>(t, Uw, H_, Ub, out);
}